// MessagePassingLayer_75703093559395
// MI455X (gfx1250) — compile-verified
//
#include <hip/hip_runtime.h>

// Problem constants (from reference): B=4, N=256, F=128, H=256, STEPS=3
#define BN 4
#define NN 256
#define FF 128
#define HH 256
#define MM (BN * NN)   /* 1024 rows */
#define NSTEPS 3

typedef __attribute__((ext_vector_type(2))) float        v2f;
typedef __attribute__((ext_vector_type(8))) float        v8f;
typedef __attribute__((ext_vector_type(4))) unsigned int v4u;
typedef __attribute__((ext_vector_type(4))) int          v4i;
typedef __attribute__((ext_vector_type(8))) int          v8i;

// ---------------------------------------------------------------------------
// TDM helper: DMA a dense [rows x K] fp32 tile (row stride = K elements) from
// global memory into LDS offset 0. Issued by the calling wave; caller must
// s_wait_tensorcnt + barrier before consuming. D# layout per CDNA5 ISA ch.8:
//   group0: count=1 | lds_addr | global_addr[56:0] | type=2
//   group1: data_size=4B (code 2), tensor_dim0=K, tensor_dim1=rows,
//           tile_dim0=K, tile_dim1=rows, tensor_dim0_stride=K
//   groups 2/3: zero (<=2D tensor)
// ---------------------------------------------------------------------------
__device__ __forceinline__ void tdm_load_tile_f32(const float* gsrc, int K, int rows)
{
  const unsigned long long ga = (unsigned long long)(__UINTPTR_TYPE__)gsrc;
  v4u g0;
  g0[0] = 1u;                                           // count=1 (valid user D#)
  g0[1] = 0u;                                           // lds_addr = 0 (sole LDS object)
  g0[2] = (unsigned)(ga & 0xFFFFFFFFu);                 // global_addr[31:0]
  g0[3] = (unsigned)((ga >> 32) & 0x01FFFFFFu) | (2u << 30); // addr[56:32] | type=2
  v8i g1;
  g1[0] = (int)(2u << 16);                              // data_size = 4 bytes
  g1[1] = (int)(((unsigned)K & 0xFFFFu) << 16);         // tensor_dim0[15:0]
  g1[2] = (int)(((unsigned)rows & 0xFFFFu) << 16);      // tensor_dim0 hi=0 | tensor_dim1 lo
  g1[3] = (int)(((unsigned)K & 0xFFFFu) << 16);         // tensor_dim1 hi=0 | tile_dim0=K
  g1[4] = rows;                                         // tile_dim1=rows, tile_dim2=0
  g1[5] = K;                                            // tensor_dim0_stride[31:0]
  g1[6] = 0;                                            // stride hi / dim1_stride lo
  g1[7] = 0;
  v4i g2 = {0, 0, 0, 0};
  v4i g3 = {0, 0, 0, 0};
#if __has_include(<hip/amd_detail/amd_gfx1250_TDM.h>)
  v8i g4 = {0, 0, 0, 0, 0, 0, 0, 0};                    // therock toolchain: 6-arg form
  __builtin_amdgcn_tensor_load_to_lds(g0, g1, g2, g3, g4, 0);
#else
  __builtin_amdgcn_tensor_load_to_lds(g0, g1, g2, g3, 0); // ROCm 7.2: 5-arg form
#endif
}

// ---------------------------------------------------------------------------
// Generic fp32 WMMA GEMM:  C[M x Nd] = A[M x K] * W[:, woff:woff+K]^T
//                                      + bias[n] * (HAS_SCALE ? rowscale[m] : 1)
// W row-major (Nd x ldw). Block = 128 threads = 4 waves; all 4 waves share one
// 16-row A tile, DMA'd into LDS by the TDM once per block; each wave computes
// one 16x16 C tile via V_WMMA_F32_16X16X4_F32. Grid: (M/16, Ntiles/4).
// Templated on bias/scale so no runtime-null loop versioning pollutes codegen.
// ---------------------------------------------------------------------------
template <bool HAS_BIAS, bool HAS_SCALE>
__global__ __launch_bounds__(128)
void wmma_gemm_f32(const float* __restrict__ A, int K,
                   const float* __restrict__ W, int ldw, int woff,
                   const float* __restrict__ bias,
                   const float* __restrict__ rowscale,
                   float* __restrict__ C, int Nd)
{
  __shared__ float sA[16 * 256];                 // 16 rows x K (K <= 256) = 16 KB max

  const int lane  = threadIdx.x & 31;
  const int wv    = threadIdx.x >> 5;
  const int mbase = blockIdx.x * 16;
  const int nbase = (blockIdx.y * 4 + wv) * 16;

  // One TDM DMA per block: A[mbase:mbase+16, 0:K] -> sA (dense 16 x K).
  if (wv == 0) {
    tdm_load_tile_f32(A + (size_t)mbase * K, K, 16);
    __builtin_amdgcn_s_wait_tensorcnt(0);        // TENSORcnt is per-wave
  }
  __syncthreads();                               // publish LDS to other 3 waves
  // The TDM write to sA is invisible to the compiler; without this clobber it
  // folds loads of the never-"stored" shared array to undef and hoists them.
  asm volatile("" ::: "memory");

  // 16x4 fp32 A fragment (ISA 7.12.2): lanes 0-15 M=lane, K={k0,k0+1};
  // lanes 16-31 M=lane-16, K={k0+2,k0+3}. B (4x16) mirrors with N across lanes.
  const int r  = lane & 15;
  const int kh = (lane >> 4) << 1;               // 0 or 2

  const float* arow = sA + (size_t)r * K;
  const float* wrow = W + (size_t)(nbase + r) * ldw + woff;

  v8f c = {0.f, 0.f, 0.f, 0.f, 0.f, 0.f, 0.f, 0.f};
  for (int k0 = 0; k0 < K; k0 += 4) {
    v2f a, b;
    a.x = arow[k0 + kh + 0];
    a.y = arow[k0 + kh + 1];
    b.x = wrow[k0 + kh + 0];
    b.y = wrow[k0 + kh + 1];
    // (neg_a, A, neg_b, B, c_mod, C, reuse_a, reuse_b)
    c = __builtin_amdgcn_wmma_f32_16x16x4_f32(false, a, false, b,
                                              (short)0, c, false, false);
  }

  // 16x16 f32 C/D layout: VGPR rr -> M = rr (lanes 0-15) / M = rr+8 (16-31).
  const int n  = lane & 15;
  const int mo = (lane >> 4) << 3;
  const float bv = HAS_BIAS ? bias[nbase + n] : 0.0f;
#pragma unroll
  for (int rr = 0; rr < 8; ++rr) {
    const int   mrow = mbase + mo + rr;
    float out = c[rr];
    if (HAS_BIAS) out += bv * (HAS_SCALE ? rowscale[mrow] : 1.0f);
    C[(size_t)mrow * Nd + nbase + n] = out;
  }
}

// ---------------------------------------------------------------------------
// Pairwise masked relu aggregation (the only O(N^2) work after pulling the
// linear W2 out of the j-sum):
//   T[b,i,h] = sum_j valid[b,i,j] * relu(psb[b,i,h] + pt[b,j,h])
//   deg[b,i] = sum_j valid[b,i,j]
// psb already contains +b1. One block per (b,i); 256 threads over h.
// ---------------------------------------------------------------------------
__global__ __launch_bounds__(256)
void pairwise_agg(const float* __restrict__ psb, const float* __restrict__ pt,
                  const float* __restrict__ adj, const unsigned char* __restrict__ mask,
                  float* __restrict__ T, float* __restrict__ deg)
{
  const int bi = blockIdx.x;          // b*N + i
  const int b  = bi >> 8;
  const int i  = bi & (NN - 1);
  const int h  = threadIdx.x;

  __shared__ float s_adj[NN];
  __shared__ float s_msk[NN];
  s_adj[h] = adj[(size_t)bi * NN + h];
  s_msk[h] = mask[b * NN + h] ? 1.0f : 0.0f;
  __syncthreads();

  const float  mi  = s_msk[i];
  const float  psv = psb[(size_t)bi * HH + h];
  const float* ptb = pt + (size_t)b * NN * HH + h;

  float acc = 0.0f, dsum = 0.0f;
  for (int j = 0; j < NN; ++j) {
    if (j + 8 < NN) __builtin_prefetch(ptb + (size_t)(j + 8) * HH, 0, 1);
    const float v  = (s_adj[j] > 0.0f) ? (mi * s_msk[j]) : 0.0f;
    const float xx = psv + ptb[(size_t)j * HH];
    acc  += v * fmaxf(xx, 0.0f);
    dsum += v;
  }
  T[(size_t)bi * HH + h] = acc;
  if (h == 0) deg[bi] = dsum;
}

// ---------------------------------------------------------------------------
// GRU gate fusion + in-place h update:
//   r = sig(gi_r+gh_r); z = sig(gi_z+gh_z); n = tanh(gi_n + r*gh_n)
//   h = ((1-z)*n + z*h) * mask
// ---------------------------------------------------------------------------
__global__ __launch_bounds__(256)
void gru_gate(const float* __restrict__ gi, const float* __restrict__ gh,
              float* __restrict__ h, const unsigned char* __restrict__ mask)
{
  const int idx = blockIdx.x * 256 + threadIdx.x;   // over M*F = 131072
  const int m   = idx >> 7;                         // / F
  const int f   = idx & (FF - 1);

  const float* gim = gi + (size_t)m * (3 * FF);
  const float* ghm = gh + (size_t)m * (3 * FF);

  const float rsum = gim[f]      + ghm[f];
  const float zsum = gim[FF + f] + ghm[FF + f];
  const float r = 1.0f / (1.0f + __expf(-rsum));
  const float z = 1.0f / (1.0f + __expf(-zsum));
  const float n = tanhf(gim[2 * FF + f] + r * ghm[2 * FF + f]);

  const float hv  = h[idx];
  const float out = (1.0f - z) * n + z * hv;
  h[idx] = mask[m] ? out : 0.0f;                    // mask flat index == m
}

// ---------------------------------------------------------------------------
extern "C" void kernel_launch(void* const* d_in, const int* in_sizes, int n_in,
                              void* d_out, int out_size, void* d_ws, size_t ws_size,
                              hipStream_t stream)
{
  const float*         x    = (const float*)d_in[0];   // (B,N,F)
  const float*         adj  = (const float*)d_in[1];   // (B,N,N)
  const unsigned char* mask = (const unsigned char*)d_in[2]; // (B,N) bool
  const float*         W1   = (const float*)d_in[3];   // (H, 2F)
  const float*         b1   = (const float*)d_in[4];   // (H,)
  const float*         W2   = (const float*)d_in[5];   // (F, H)
  const float*         b2   = (const float*)d_in[6];   // (F,)
  const float*         W_ih = (const float*)d_in[7];   // (3F, F)
  const float*         W_hh = (const float*)d_in[8];   // (3F, F)
  const float*         b_ih = (const float*)d_in[9];   // (3F,)
  const float*         b_hh = (const float*)d_in[10];  // (3F,)

  // Workspace carve-up (floats): ~7.3 MB total, fully L2-resident.
  float* ws  = (float*)d_ws;
  float* h   = ws;                        // M*F    = 131072
  float* ps  = h   + (size_t)MM * FF;     // M*H    = 262144 (holds ps + b1)
  float* pt  = ps  + (size_t)MM * HH;     // M*H
  float* T   = pt  + (size_t)MM * HH;     // M*H
  float* deg = T   + (size_t)MM * HH;     // M      = 1024
  float* agg = deg + MM;                  // M*F
  float* gi  = agg + (size_t)MM * FF;     // M*3F   = 393216
  float* gh  = gi  + (size_t)MM * 3 * FF; // M*3F

  // h = x
  hipMemcpyAsync(h, x, (size_t)MM * FF * sizeof(float),
                 hipMemcpyDeviceToDevice, stream);

  const dim3 blk(128);
  for (int step = 0; step < NSTEPS; ++step) {
    // ps = h @ Ws^T + b1   (Ws = W1[:, :F]);  pt = h @ Wt^T  (Wt = W1[:, F:])
    wmma_gemm_f32<true, false><<<dim3(MM / 16, (HH / 16) / 4), blk, 0, stream>>>(
        h, FF, W1, 2 * FF, 0, b1, nullptr, ps, HH);
    wmma_gemm_f32<false, false><<<dim3(MM / 16, (HH / 16) / 4), blk, 0, stream>>>(
        h, FF, W1, 2 * FF, FF, nullptr, nullptr, pt, HH);

    // T[b,i,h] = sum_j valid * relu(ps+pt);  deg[b,i] = sum_j valid
    pairwise_agg<<<MM, NN, 0, stream>>>(ps, pt, adj, mask, T, deg);

    // agg = T @ W2^T + b2 * deg
    wmma_gemm_f32<true, true><<<dim3(MM / 16, (FF / 16) / 4), blk, 0, stream>>>(
        T, HH, W2, HH, 0, b2, deg, agg, FF);

    // gi = agg @ W_ih^T + b_ih ;  gh = h @ W_hh^T + b_hh
    wmma_gemm_f32<true, false><<<dim3(MM / 16, (3 * FF / 16) / 4), blk, 0, stream>>>(
        agg, FF, W_ih, FF, 0, b_ih, nullptr, gi, 3 * FF);
    wmma_gemm_f32<true, false><<<dim3(MM / 16, (3 * FF / 16) / 4), blk, 0, stream>>>(
        h, FF, W_hh, FF, 0, b_hh, nullptr, gh, 3 * FF);

    // GRU gates, in-place h update with mask
    gru_gate<<<(MM * FF) / 256, 256, 0, stream>>>(gi, gh, h, mask);
  }

  hipMemcpyAsync(d_out, h, (size_t)MM * FF * sizeof(float),
                 hipMemcpyDeviceToDevice, stream);
}